// SetCriterion_43353399886185
// MI455X (gfx1250) — compile-verified
//
#include <hip/hip_runtime.h>

#define B_  64
#define Q_  2048
#define CE_ 64
#define CS_ 32
#define NG_ 100

#define GB_       512                  // blocks per head in the streaming kernel
#define MAIN_OFF_ 0                    // ws[0 .. 7*GB_-1]      : per-block head partials
#define CORR_OFF_ (7 * GB_)           // ws[7*GB_ .. +B_*8-1]  : per-batch corr partials

typedef float v2f __attribute__((ext_vector_type(2)));
typedef float v8f __attribute__((ext_vector_type(8)));

// g(x) = softplus(x) * sigmoid(x)^2   (focal loss term for target==0, unscaled)
__device__ __forceinline__ float focal_g(float x) {
    float t  = __expf(-fabsf(x));          // e^{-|x|} in (0,1]
    float r  = 1.0f / (1.0f + t);
    float lp = __logf(1.0f + t);           // softplus(-|x|)
    float sp = lp + fmaxf(x, 0.0f);        // softplus(x), stable
    float p  = (x >= 0.0f) ? r : t * r;    // sigmoid(x)
    return sp * p * p;
}

// delta at a scatter point: 0.25*g(-x) - 0.75*g(x)
__device__ __forceinline__ float focal_delta(float x) {
    float t  = __expf(-fabsf(x));
    float r  = 1.0f / (1.0f + t);
    float lp = __logf(1.0f + t);
    float sp_p = lp + fmaxf(x, 0.0f);      // softplus(x)
    float sp_n = lp + fmaxf(-x, 0.0f);     // softplus(-x)
    float p = (x >= 0.0f) ? r : t * r;     // sigmoid(x)
    float q = 1.0f - p;                    // sigmoid(-x)
    return 0.25f * sp_n * q * q - 0.75f * sp_p * p * p;
}

// Wave32 reduction via V_WMMA_F32_16X16X4_F32: A = per-lane partials (16x4),
// B = all-ones (4x16) => D rows are K-sums. Lanes 0-15 end with sum of rows
// 0-7, lanes 16-31 with rows 8-15; one shfl_xor(16) yields the full wave sum.
// Must be called with uniform control flow (EXEC all ones).
__device__ __forceinline__ float wave_reduce_wmma(float a0, float a1) {
    v2f a; a.x = a0;   a.y = a1;
    v2f b; b.x = 1.0f; b.y = 1.0f;
    v8f c = {};
    v8f d = __builtin_amdgcn_wmma_f32_16x16x4_f32(false, a, false, b,
                                                  (short)0, c, false, false);
    float s = d[0] + d[1] + d[2] + d[3] + d[4] + d[5] + d[6] + d[7];
    s += __shfl_xor(s, 16, 32);
    return s;   // full wave sum, broadcast to all lanes
}

// grid = (GB_, 7): blockIdx.y selects the head. Streams the logit tensor as
// float4 with a x2-unrolled grid-stride loop (two in-flight B128 loads),
// WMMA wave reduce, ordered cross-wave sum, one partial per block.
__global__ void __launch_bounds__(256)
focal_sum_kernel(const float* __restrict__ h0, const float* __restrict__ h1,
                 const float* __restrict__ h2, const float* __restrict__ h3,
                 const float* __restrict__ h4, const float* __restrict__ h5,
                 const float* __restrict__ h6, float* __restrict__ ws) {
    const int head = blockIdx.y;
    const float* src; uint32_t n;
    switch (head) {
        case 0: src = h0; n = B_ * Q_ * CE_; break;
        case 1: src = h1; n = B_ * Q_ * CE_; break;
        case 2: src = h2; n = B_ * Q_ * CE_; break;
        case 3: src = h3; n = B_ * Q_ * CS_; break;
        case 4: src = h4; n = B_ * Q_ * CS_; break;
        case 5: src = h5; n = B_ * Q_ * CS_; break;
        default: src = h6; n = B_ * Q_ * CS_; break;
    }
    const float4* s4 = (const float4*)src;
    const uint32_t n4     = n >> 2;                         // multiples of 4
    const uint32_t stride = gridDim.x * blockDim.x;         // 131072
    uint32_t i = blockIdx.x * blockDim.x + threadIdx.x;

    float a0 = 0.0f, a1 = 0.0f, b0 = 0.0f, b1 = 0.0f;
    for (; i + stride < n4; i += 2 * stride) {
        __builtin_prefetch(s4 + i + 4 * stride, 0, 0);      // global_prefetch_b8
        float4 v = s4[i];
        float4 w = s4[i + stride];
        a0 += focal_g(v.x) + focal_g(v.y);
        a1 += focal_g(v.z) + focal_g(v.w);
        b0 += focal_g(w.x) + focal_g(w.y);
        b1 += focal_g(w.z) + focal_g(w.w);
    }
    if (i < n4) {
        float4 v = s4[i];
        a0 += focal_g(v.x) + focal_g(v.y);
        a1 += focal_g(v.z) + focal_g(v.w);
    }

    // uniform control flow: every lane executes the WMMA
    const float wsum = wave_reduce_wmma(a0 + b0, a1 + b1);

    __shared__ float s_wave[8];                 // 256 threads = 8 waves
    const int wid  = threadIdx.x >> 5;
    const int lane = threadIdx.x & 31;
    if (lane == 0) s_wave[wid] = wsum;
    __syncthreads();
    if (threadIdx.x == 0) {                     // ordered, deterministic
        float s = s_wave[0];
        #pragma unroll
        for (int w = 1; w < 8; ++w) s += s_wave[w];
        ws[MAIN_OFF_ + head * GB_ + blockIdx.x] = s;
    }
}

// One block (4 waves) per batch. Computes the L1 point loss and the per-head
// sparse focal corrections with exact first-occurrence dedup of (q, cls)
// scatters; reduces all 8 accumulators with the WMMA wave reducer.
__global__ void __launch_bounds__(128)
corr_kernel(const float* __restrict__ pred_points, const float* __restrict__ points_gt,
            const int* __restrict__ obj_idx, const int* __restrict__ gt_idx,
            const float* l0, const float* l1, const float* l2, const float* l3,
            const float* l4, const float* l5, const float* l6,
            const int* g0, const int* g1, const int* g2, const int* g3,
            const int* g4, const int* g5, const int* g6,
            float* __restrict__ ws) {
    const int b = blockIdx.x;
    const int t = threadIdx.x;

    const float* logits[7] = {l0, l1, l2, l3, l4, l5, l6};
    const int*   gts[7]    = {g0, g1, g2, g3, g4, g5, g6};
    const int    cdim[7]   = {CE_, CE_, CE_, CS_, CS_, CS_, CS_};

    __shared__ int   s_obj[NG_];
    __shared__ int   s_gti[NG_];
    __shared__ int   s_cls[NG_];
    __shared__ float s_wave[4][8];              // [wave][accumulator]

    if (t < NG_) {
        s_obj[t] = obj_idx[b * NG_ + t];
        s_gti[t] = gt_idx[b * NG_ + t];
    }
    __syncthreads();

    float my[8];
    #pragma unroll
    for (int k = 0; k < 8; ++k) my[k] = 0.0f;

    // L1 point loss (pure gather, no dedup)
    if (t < NG_) {
        const int q = s_obj[t];
        const int g = s_gti[t];
        const float px = pred_points[((size_t)b * Q_ + q) * 2 + 0];
        const float py = pred_points[((size_t)b * Q_ + q) * 2 + 1];
        const float gx = points_gt[((size_t)b * NG_ + g) * 2 + 0];
        const float gy = points_gt[((size_t)b * NG_ + g) * 2 + 1];
        my[0] = fabsf(px - gx) + fabsf(py - gy);
    }

    for (int h = 0; h < 7; ++h) {
        __syncthreads();
        if (t < NG_) s_cls[t] = gts[h][b * NG_ + s_gti[t]];
        __syncthreads();
        if (t < NG_) {
            const int c = s_cls[t];
            const int q = s_obj[t];
            bool first = true;                  // .set() semantics: dedup scatters
            for (int j = 0; j < t; ++j)
                if (s_obj[j] == q && s_cls[j] == c) { first = false; break; }
            if (first) {
                const float x = logits[h][((size_t)b * Q_ + q) * cdim[h] + c];
                my[1 + h] = focal_delta(x);
            }
        }
    }

    // 8 WMMA wave reductions, all unconditional (EXEC all ones)
    const int wid  = t >> 5;
    const int lane = t & 31;
    #pragma unroll
    for (int k = 0; k < 8; ++k) {
        const float wsum = wave_reduce_wmma(my[k], 0.0f);
        if (lane == 0) s_wave[wid][k] = wsum;
    }
    __syncthreads();
    if (t < 8) {                                // ordered, deterministic
        float s = s_wave[0][t];
        #pragma unroll
        for (int w = 1; w < 4; ++w) s += s_wave[w][t];
        ws[CORR_OFF_ + b * 8 + t] = s;
    }
}

// 8 blocks: block 0 -> point loss; blocks 1..7 -> head h = k-1.
// Fixed-shape tree reductions: deterministic across replays.
__global__ void __launch_bounds__(256)
finalize_kernel(const float* __restrict__ ws, float* __restrict__ out) {
    __shared__ float red[256];
    const int k = blockIdx.x;
    const int t = threadIdx.x;
    const float inv_np = 1.0f / (float)(B_ * NG_);

    if (k == 0) {
        red[t] = (t < B_) ? ws[CORR_OFF_ + t * 8 + 0] : 0.0f;
        __syncthreads();
        for (int s = 128; s > 0; s >>= 1) {
            if (t < s) red[t] += red[t + s];
            __syncthreads();
        }
        if (t == 0) out[0] = red[0] * inv_np;
    } else {
        const int h = k - 1;
        // main streaming sum over GB_ = 512 block partials
        red[t] = ws[MAIN_OFF_ + h * GB_ + t] + ws[MAIN_OFF_ + h * GB_ + t + 256];
        __syncthreads();
        for (int s = 128; s > 0; s >>= 1) {
            if (t < s) red[t] += red[t + s];
            __syncthreads();
        }
        const float S = red[0];
        __syncthreads();
        // sparse correction sum over B_ = 64 batch partials
        red[t] = (t < B_) ? ws[CORR_OFF_ + t * 8 + k] : 0.0f;
        __syncthreads();
        for (int s = 128; s > 0; s >>= 1) {
            if (t < s) red[t] += red[t + s];
            __syncthreads();
        }
        if (t == 0) out[k] = (0.75f * S + red[0]) * inv_np;
    }
}

extern "C" void kernel_launch(void* const* d_in, const int* in_sizes, int n_in,
                              void* d_out, int out_size, void* d_ws, size_t ws_size,
                              hipStream_t stream) {
    const float* pred_points = (const float*)d_in[0];
    const float* pe  = (const float*)d_in[1];
    const float* ple = (const float*)d_in[2];
    const float* pte = (const float*)d_in[3];
    const float* plu = (const float*)d_in[4];
    const float* pru = (const float*)d_in[5];
    const float* prd = (const float*)d_in[6];
    const float* pld = (const float*)d_in[7];
    const float* points_gt = (const float*)d_in[8];
    const int* obj_idx = (const int*)d_in[9];
    const int* gt_idx  = (const int*)d_in[10];
    const int* eg  = (const int*)d_in[11];
    const int* leg = (const int*)d_in[12];
    const int* teg = (const int*)d_in[13];
    const int* lug = (const int*)d_in[14];
    const int* rug = (const int*)d_in[15];
    const int* rdg = (const int*)d_in[16];
    const int* ldg = (const int*)d_in[17];

    float* ws  = (float*)d_ws;
    float* out = (float*)d_out;

    dim3 grid(GB_, 7);
    focal_sum_kernel<<<grid, 256, 0, stream>>>(pe, ple, pte, plu, pru, prd, pld, ws);

    corr_kernel<<<B_, 128, 0, stream>>>(pred_points, points_gt, obj_idx, gt_idx,
                                        pe, ple, pte, plu, pru, prd, pld,
                                        eg, leg, teg, lug, rug, rdg, ldg, ws);

    finalize_kernel<<<8, 256, 0, stream>>>(ws, out);
}